// Focalloss_exphm_and_sigma_KL_divergence_79250736546225
// MI455X (gfx1250) — compile-verified
//
#include <hip/hip_runtime.h>
#include <hip/hip_bf16.h>

typedef __attribute__((ext_vector_type(2))) float v2f;
typedef __attribute__((ext_vector_type(4))) float v4f;
typedef __attribute__((ext_vector_type(8))) float v8f;

#define RR 2
#define BB 8
#define CC 80
#define HH 128
#define WW 128
#define OO 128
#define NTOT (BB*CC*HH*WW)   // 10,485,760
#define NOBJ (BB*CC*OO)      // 81,920

// log(sigmoid(v)) = -log(1 + exp(-v))
__device__ __forceinline__ float log_sigmoid(float v) {
    return -__logf(1.0f + __expf(-v));
}

// PINV = (M^T M)^{-1} M^T for M = [1, x, y, x^2, y^2] on the 5x5 grid.
// Closed form (block-diagonal M^T M):
//   PINV[0][p] = 27/175 - (x^2+y^2)/35
//   PINV[1][p] = x/50            PINV[2][p] = y/50
//   PINV[3][p] = -1/35 + x^2/70  PINV[4][p] = -1/35 + y^2/70
constexpr float pinv_c(int k, int p) {
    if (p >= 25 || k >= 5) return 0.0f;
    const float x = (float)(p % 5 - 2);
    const float y = (float)(p / 5 - 2);
    return k == 0 ? 27.0f/175.0f - (x*x + y*y)/35.0f
         : k == 1 ? x/50.0f
         : k == 2 ? y/50.0f
         : k == 3 ? -1.0f/35.0f + x*x/70.0f
         :          -1.0f/35.0f + y*y/70.0f;
}

// B-fragment table for V_WMMA_F32_16X16X4_F32 (B is 4x16 = K x N):
// lane l -> N = l&15; K sub-row = (l>>4)*2 + vgpr.  Indexed [chunk][vgpr][lane].
struct BTabT { float v[7][2][32]; };
constexpr BTabT make_btab() {
    BTabT t{};
    for (int c = 0; c < 7; ++c)
        for (int vv = 0; vv < 2; ++vv)
            for (int l = 0; l < 32; ++l) {
                const int n = l & 15, half = l >> 4;
                const int p = 4*c + 2*half + vv;   // global K == patch index
                t.v[c][vv][l] = pinv_c(n, p);
            }
    return t;
}
__device__ __constant__ BTabT BTAB = make_btab();

__global__ void k_init(float* ws) {
    if (threadIdx.x < 4) ws[threadIdx.x] = 0.0f;
}

// ---------------- focal loss: memory-bound reduction pass ----------------
__global__ __launch_bounds__(256) void k_focal(const float* __restrict__ hm_out,
                                               const float* __restrict__ hm_gt,
                                               const float* __restrict__ hm_mask,
                                               float* __restrict__ ws) {
    const v4f* __restrict__ ho = (const v4f*)hm_out;
    const v4f* __restrict__ hg = (const v4f*)hm_gt;
    const v4f* __restrict__ hk = (const v4f*)hm_mask;
    float pl = 0.0f, nl = 0.0f, np = 0.0f;
    for (int i = blockIdx.x * blockDim.x + threadIdx.x; i < NTOT / 4;
         i += gridDim.x * blockDim.x) {
        const v4f v4 = ho[i];                                // RT: reused by k_fit
        const v4f g4 = __builtin_nontemporal_load(&hg[i]);   // single-use stream
        const v4f m4 = __builtin_nontemporal_load(&hk[i]);   // single-use stream
#pragma unroll
        for (int j = 0; j < 4; ++j) {
            const float v = v4[j];
            const float s = 1.0f / (1.0f + __expf(-v));
            const float pred = fminf(fmaxf(s, 1e-6f), 1.0f - 1e-6f);
            const float gt = g4[j] * m4[j];
            const float omp_ = 1.0f - pred;
            const float lp = __logf(pred);
            const float ln = __logf(omp_);
            const bool pos = (gt == 1.0f);
            const bool neg = (gt < 1.0f);
            const float t = 1.0f - gt;
            const float t2 = t * t;
            pl += pos ? lp * omp_ * omp_ : 0.0f;
            np += pos ? 1.0f : 0.0f;
            nl += neg ? ln * pred * pred * t2 * t2 : 0.0f;
        }
    }
    for (int off = 16; off; off >>= 1) {
        pl += __shfl_xor(pl, off, 32);
        nl += __shfl_xor(nl, off, 32);
        np += __shfl_xor(np, off, 32);
    }
    __shared__ float red[3][8];
    const int lane = threadIdx.x & 31, wid = threadIdx.x >> 5;
    if (lane == 0) { red[0][wid] = pl; red[1][wid] = nl; red[2][wid] = np; }
    __syncthreads();
    if (threadIdx.x == 0) {
        float a = 0.0f, b = 0.0f, c = 0.0f;
        for (int w = 0; w < 8; ++w) { a += red[0][w]; b += red[1][w]; c += red[2][w]; }
        atomicAdd(&ws[1], a);
        atomicAdd(&ws[2], b);
        atomicAdd(&ws[3], c);
    }
}

// ------------- Gaussian fit (WMMA f32 16x16x4) + KL divergence -------------
// One wave = 16 consecutive objects (all share one (b,c) since O=128).
// logz[16,25] x PINV^T[25,5] -> 7 chained V_WMMA_F32_16X16X4_F32 (K padded to 28).
// A 16x4 f32: lanes 0-15 hold K={0,1} in vgprs {0,1}; lanes 16-31 K={2,3}.
// D 16x16 f32: vgpr j, lanes 0-15 -> M=j, lanes 16-31 -> M=j+8; N=lane%16.
__global__ __launch_bounds__(256) void k_fit(const float* __restrict__ hm_out,
                                             const int* __restrict__ ct_ind,
                                             const float* __restrict__ sigma_wh,
                                             const int* __restrict__ swh_mask,
                                             float* __restrict__ ws) {
    __shared__ float coef[8][4][16];   // [wave][bx,by,ax,ay][object]
    const int lane = threadIdx.x & 31;
    const int wid  = threadIdx.x >> 5;
    const int wave_g = blockIdx.x * 8 + wid;        // 5120 waves total
    const int m    = lane & 15;                     // object-in-wave / N column
    const int half = lane >> 4;                     // K sub-rows {0,1} vs {2,3}
    const int bc   = wave_g >> 3;                   // b*C + c (wave-uniform)
    const int o    = ((wave_g & 7) << 4) + m;

    const int ci = (bc * OO + o) * 2;
    int cx = ct_ind[ci + 0]; cx = cx < RR ? RR : (cx > WW - 1 - RR ? WW - 1 - RR : cx);
    int cy = ct_ind[ci + 1]; cy = cy < RR ? RR : (cy > HH - 1 - RR ? HH - 1 - RR : cy);
    const float* __restrict__ hm = hm_out + (size_t)bc * (HH * WW);
    const int base = cy * WW + cx;

    v8f acc = {};
#pragma unroll
    for (int c = 0; c < 7; ++c) {
        v2f a, bf;
#pragma unroll
        for (int v = 0; v < 2; ++v) {
            const int KL = 4*c + v;        // K if half==0
            const int KH = 4*c + 2 + v;    // K if half==1
            float av = 0.0f;
            if (KL < 25) {                 // compile-time (KH<25 => KL<25)
                const int pL = KL;
                const int pH = (KH < 25) ? KH : 12;          // safe dummy addr
                const int offL = (pL/5 - 2) * WW + (pL%5 - 2);
                const int offH = (pH/5 - 2) * WW + (pH%5 - 2);
                const int off  = half ? offH : offL;         // v_cndmask, no branch
                const float ls = log_sigmoid(hm[base + off]);
                const bool ok  = half ? (KH < 25) : true;
                av = ok ? ls : 0.0f;
            }
            a[v]  = av;
            bf[v] = BTAB.v[c][v][lane];    // constant-cache broadcast, branch-free
        }
        acc = __builtin_amdgcn_wmma_f32_16x16x4_f32(
            /*neg_a=*/false, a, /*neg_b=*/false, bf,
            /*c_mod=*/(short)0, acc, /*reuse_a=*/false, /*reuse_b=*/false);
    }

    // scatter coefficient columns k=1..4 of D into LDS by object
    if (m >= 1 && m <= 4) {
#pragma unroll
        for (int j = 0; j < 8; ++j)
            coef[wid][m - 1][j + half * 8] = acc[j];
    }
    __syncthreads();

    // cnt = sum(sigmawh_mask[b,c,:]) — wave-uniform reduction over 128 ints
    const int* __restrict__ mk = swh_mask + bc * OO;
    int cnt = 0;
#pragma unroll
    for (int i = 0; i < 4; ++i) cnt += mk[lane * 4 + i];
    for (int off = 16; off; off >>= 1) cnt += __shfl_xor(cnt, off, 32);

    float kl = 0.0f;
    if (lane < 16) {
        const float bx = coef[wid][0][lane];
        const float by = coef[wid][1][lane];
        const float ax = coef[wid][2][lane];
        const float ay = coef[wid][3][lane];
        const float axs = (fabsf(ax) < 1e-12f) ? -1.0f : ax;
        const float ays = (fabsf(ay) < 1e-12f) ? -1.0f : ay;
        const float sw2 = -0.5f / axs;
        const float sh2 = -0.5f / ays;
        const float mu_w = -bx / (2.0f * axs);
        const float mu_h = -by / (2.0f * ays);
        const int oo = ((wave_g & 7) << 4) + lane;
        const bool valid = (mk[oo] == 1) && (oo < cnt) && (sw2 > 0.0f) && (sh2 > 0.0f);
        const float sgw = sigma_wh[(bc * OO + oo) * 2 + 0];
        const float sgh = sigma_wh[(bc * OO + oo) * 2 + 1];
        const float osw = valid ? sw2 : 1.0f;
        const float osh = valid ? sh2 : 1.0f;
        const float omw = valid ? mu_w : 0.0f;
        const float omh = valid ? mu_h : 0.0f;
        const float gw  = valid ? sgw : 1.0f;
        const float gh  = valid ? sgh : 1.0f;
        const float sgw2 = gw * gw, sgh2 = gh * gh;
        const float klv = 0.5f * __logf(sgw2 * sgh2 / (osw * osh)) - 1.0f
                        + 0.5f * (osw / sgw2 + osh / sgh2)
                        + 0.5f * (omw * omw / sgw2 + omh * omh / sgh2);
        kl = valid ? klv : 0.0f;
    }
    for (int off = 16; off; off >>= 1) kl += __shfl_xor(kl, off, 32);
    if (lane == 0) atomicAdd(&ws[0], kl);
}

__global__ void k_fin(const float* __restrict__ ws, float* __restrict__ out) {
    if (threadIdx.x == 0 && blockIdx.x == 0) {
        const float np = ws[3];
        const float focal = (np == 0.0f) ? -ws[2] : -(ws[1] + ws[2]) / np;
        out[0] = ws[0];    // kl_sum
        out[1] = focal;    // focal
    }
}

extern "C" void kernel_launch(void* const* d_in, const int* in_sizes, int n_in,
                              void* d_out, int out_size, void* d_ws, size_t ws_size,
                              hipStream_t stream) {
    const float* hm_out   = (const float*)d_in[0];
    const float* hm_gt    = (const float*)d_in[1];
    const int*   ct_ind   = (const int*)d_in[2];
    const float* sigma_wh = (const float*)d_in[3];
    const float* hm_mask  = (const float*)d_in[4];
    const int*   swh_mask = (const int*)d_in[5];
    float* out = (float*)d_out;
    float* ws  = (float*)d_ws;   // ws[0]=kl, ws[1]=pos, ws[2]=neg, ws[3]=num_pos

    hipLaunchKernelGGL(k_init, dim3(1), dim3(32), 0, stream, ws);
    hipLaunchKernelGGL(k_focal, dim3(2048), dim3(256), 0, stream,
                       hm_out, hm_gt, hm_mask, ws);
    hipLaunchKernelGGL(k_fit, dim3(NOBJ / 128), dim3(256), 0, stream,
                       hm_out, ct_ind, sigma_wh, swh_mask, ws);
    hipLaunchKernelGGL(k_fin, dim3(1), dim3(32), 0, stream, ws, out);
}